// LIFSpike_14482629722900
// MI455X (gfx1250) — compile-verified
//
#include <hip/hip_runtime.h>
#include <cstdint>

// LIF ternary-spike scan over the time axis (8 steps, innermost/contiguous).
// Memory-bound streaming kernel using the CDNA5 async-to-LDS path with a
// double-buffered software pipeline (ASYNCcnt), wave32-friendly b128 accesses,
// and non-temporal output stores.

typedef float v4f __attribute__((ext_vector_type(4)));

#define TAU_F 0.25f
#define VTH_F 0.3f
#define STEPS 8
#define BLOCK 256
#define TILE_BYTES (BLOCK * STEPS * 4) /* 8192 B per tile buffer */

__global__ __launch_bounds__(BLOCK) void lif_scan_async(
    const float* __restrict__ x, float* __restrict__ out, int n_neurons) {
  // 16 KB double buffer: each lane owns a private 32B slice of each buffer.
  __shared__ float tile[2 * BLOCK * STEPS];

  const int lane = threadIdx.x;
  const int n_tiles = (n_neurons + BLOCK - 1) / BLOCK;
  const int stride = gridDim.x;

  // Hardware LDS byte address of this lane's slices (low 32 bits of the
  // generic shared-aperture address are the LDS offset).
  const unsigned lds_base = (unsigned)(uintptr_t)(&tile[0]);
  const unsigned my_lds0 = lds_base + (unsigned)lane * (STEPS * 4);
  const unsigned my_lds1 = my_lds0 + TILE_BYTES;

  int t = blockIdx.x;
  if (t >= n_tiles) return;

  // Prologue: issue async copy of the first tile into buffer 0.
  {
    int neuron = t * BLOCK + lane;
    int nc = neuron < n_neurons ? neuron : (n_neurons - 1);
    const float* g = x + (size_t)nc * STEPS;
    asm volatile(
        "global_load_async_to_lds_b128 %0, %1, off\n\t"
        "global_load_async_to_lds_b128 %0, %1, off offset:16"
        :: "v"(my_lds0), "v"(g)
        : "memory");
  }

  int parity = 0;
  for (; t < n_tiles; t += stride) {
    const int tn = t + stride;
    const bool more = tn < n_tiles;

    if (more) {
      // Issue next tile into the other buffer (its previous contents were
      // fully consumed last iteration; DScnt waits precede this point).
      int neuron = tn * BLOCK + lane;
      int nc = neuron < n_neurons ? neuron : (n_neurons - 1);
      const float* g = x + (size_t)nc * STEPS;
      unsigned dst = parity ? my_lds0 : my_lds1;
      asm volatile(
          "global_load_async_to_lds_b128 %0, %1, off\n\t"
          "global_load_async_to_lds_b128 %0, %1, off offset:16"
          :: "v"(dst), "v"(g)
          : "memory");
      // 2 newest async ops may remain outstanding; current tile's 2 are done.
      asm volatile("s_wait_asynccnt 0x2" ::: "memory");
    } else {
      asm volatile("s_wait_asynccnt 0x0" ::: "memory");
    }

    // Read this lane's 8 time steps from LDS (2x ds_load_b128).
    const unsigned src_off = (parity ? my_lds1 : my_lds0) - lds_base;
    const v4f* lp = (const v4f*)((const char*)tile + src_off);
    v4f a = lp[0];
    v4f b = lp[1];

    float w[STEPS] = {a.x, a.y, a.z, a.w, b.x, b.y, b.z, b.w};
    float r[STEPS];
    float u = 0.0f, o = 0.0f;
#pragma unroll
    for (int s = 0; s < STEPS; ++s) {
      // u = TAU*u*(1-|o|) + w ; o in {-1,0,1} exactly, so (1-|o|) is 0 or 1.
      u = (o == 0.0f) ? __builtin_fmaf(TAU_F, u, w[s]) : w[s];
      float pos = (u > VTH_F) ? 1.0f : 0.0f;
      float neg = (u < -VTH_F) ? 1.0f : 0.0f;
      o = pos - neg;
      r[s] = o;
    }

    const int neuron = t * BLOCK + lane;
    if (neuron < n_neurons) {
      v4f r0 = {r[0], r[1], r[2], r[3]};
      v4f r1 = {r[4], r[5], r[6], r[7]};
      v4f* op = (v4f*)(out + (size_t)neuron * STEPS);
      // Output is write-once, never re-read: non-temporal to spare L2.
      __builtin_nontemporal_store(r0, op);
      __builtin_nontemporal_store(r1, op + 1);
    }

    parity ^= 1;
  }
}

extern "C" void kernel_launch(void* const* d_in, const int* in_sizes, int n_in,
                              void* d_out, int out_size, void* d_ws, size_t ws_size,
                              hipStream_t stream) {
  (void)n_in; (void)out_size; (void)d_ws; (void)ws_size;
  const float* x = (const float*)d_in[0];
  float* out = (float*)d_out;

  const int n_total = in_sizes[0];          // 33,554,432
  const int n_neurons = n_total / STEPS;    // 4,194,304
  const int n_tiles = (n_neurons + BLOCK - 1) / BLOCK;

  // ~4 tiles per block so the async double buffer actually pipelines,
  // while still launching ~4096 blocks (32K wave32s) to saturate HBM.
  int grid = (n_tiles + 3) / 4;
  if (grid < 1) grid = 1;
  if (grid > n_tiles) grid = n_tiles;

  lif_scan_async<<<grid, BLOCK, 0, stream>>>(x, out, n_neurons);
}